// MLF_unit_54374285968046
// MI455X (gfx1250) — compile-verified
//
#include <hip/hip_runtime.h>

// MLF multi-level spiking scan, T=5, tau=0.5, thresholds {1,2,4}.
// Memory-bound streaming kernel (~168 MB touched once -> ~7.2 us floor at
// 23.3 TB/s; AI ~1.9 FLOP/B so matrix ops are inapplicable). Each thread
// processes 4 consecutive spatial lanes (b128 path) across all 5 timesteps.
// Inputs for all timesteps are fetched up-front (independent addresses ->
// 5 in-flight global_load_b128 per thread), the sequential membrane update
// runs in registers, results stored as 5 b128 non-temporal stores.

#define MLF_TIME 5

typedef __attribute__((ext_vector_type(4))) float v4f;

__global__ __launch_bounds__(256) void mlf_unit_kernel(
    const v4f* __restrict__ x,   // [T * n4] float4
    v4f* __restrict__ o,         // [T * n4] float4
    int n4)                      // float4 elements per timestep
{
    const int i = blockIdx.x * blockDim.x + threadIdx.x;
    if (i >= n4) return;

    const float TAU  = 0.5f;
    const float VTH1 = 1.0f, VTH2 = 2.0f, VTH3 = 4.0f;

    // Fetch all timesteps (independent) -> 5 outstanding global_load_b128.
    v4f xv[MLF_TIME];
#pragma unroll
    for (int t = 0; t < MLF_TIME; ++t) {
        xv[t] = __builtin_nontemporal_load(&x[(size_t)t * (size_t)n4 + (size_t)i]);
    }

    float u1[4] = {0.f, 0.f, 0.f, 0.f};
    float u2[4] = {0.f, 0.f, 0.f, 0.f};
    float u3[4] = {0.f, 0.f, 0.f, 0.f};

    v4f ov[MLF_TIME];

#pragma unroll
    for (int t = 0; t < MLF_TIME; ++t) {
#pragma unroll
        for (int c = 0; c < 4; ++c) {
            const float xt = xv[t][c];
            // reset-on-spike leaky integrate: u = (u>Vth ? 0 : tau*u) + x_t
            u1[c] = (u1[c] > VTH1 ? 0.0f : TAU * u1[c]) + xt;
            u2[c] = (u2[c] > VTH2 ? 0.0f : TAU * u2[c]) + xt;
            u3[c] = (u3[c] > VTH3 ? 0.0f : TAU * u3[c]) + xt;
            // multi-level output: sum of post-update spike indicators
            ov[t][c] = (u1[c] > VTH1 ? 1.0f : 0.0f)
                     + (u2[c] > VTH2 ? 1.0f : 0.0f)
                     + (u3[c] > VTH3 ? 1.0f : 0.0f);
        }
    }

#pragma unroll
    for (int t = 0; t < MLF_TIME; ++t) {
        __builtin_nontemporal_store(ov[t], &o[(size_t)t * (size_t)n4 + (size_t)i]);
    }
}

extern "C" void kernel_launch(void* const* d_in, const int* in_sizes, int n_in,
                              void* d_out, int out_size, void* d_ws, size_t ws_size,
                              hipStream_t stream)
{
    (void)n_in; (void)out_size; (void)d_ws; (void)ws_size;

    const float* x = (const float*)d_in[0];
    float*       o = (float*)d_out;

    const long long total     = in_sizes[0];          // 5*32*128*32*32 = 20,971,520
    const long long n_spatial = total / MLF_TIME;     // 4,194,304 (divisible by 4)
    const int       n4        = (int)(n_spatial / 4); // 1,048,576 float4 per timestep

    const int block = 256;
    const int grid  = (n4 + block - 1) / block;       // 4096 blocks

    mlf_unit_kernel<<<grid, block, 0, stream>>>(
        reinterpret_cast<const v4f*>(x),
        reinterpret_cast<v4f*>(o),
        n4);
}